// LightGCN_20779051778107
// MI455X (gfx1250) — compile-verified
//
#include <hip/hip_runtime.h>
#include <hip/hip_bf16.h>

typedef __attribute__((ext_vector_type(2))) float v2f;
typedef __attribute__((ext_vector_type(8))) float v8f;

#define D 64
#define QUADS 16   // D/4

// ---------------- zero a float buffer (float4 stores) ----------------
__global__ void zero_f32(float* __restrict__ p, int n4) {
    int i = blockIdx.x * blockDim.x + threadIdx.x;
    if (i < n4) {
        float4 z = make_float4(0.f, 0.f, 0.f, 0.f);
        ((float4*)p)[i] = z;
    }
}

__global__ void zero_scalars(float* s) { s[0] = 0.f; s[1] = 0.f; }

// ------- layer-0 gathers: acc = table[idx], plus reg-term squares -------
__global__ void init_acc(const float* __restrict__ utab,
                         const float* __restrict__ itab,
                         const int* __restrict__ user,
                         const int* __restrict__ pos,
                         const int* __restrict__ neg,
                         float* __restrict__ u_acc,
                         float* __restrict__ p_acc,
                         float* __restrict__ n_acc,
                         float* __restrict__ sq_acc, int B) {
    int t = blockIdx.x * blockDim.x + threadIdx.x;  // B*QUADS threads
    if (t >= B * QUADS) return;
    int b = t >> 4, q = (t & 15) * 4;
    float4 u = *(const float4*)(utab + (size_t)user[b] * D + q);
    float4 p = *(const float4*)(itab + (size_t)pos[b]  * D + q);
    float4 n = *(const float4*)(itab + (size_t)neg[b]  * D + q);
    *(float4*)(u_acc + (size_t)b * D + q) = u;
    *(float4*)(p_acc + (size_t)b * D + q) = p;
    *(float4*)(n_acc + (size_t)b * D + q) = n;
    float sq = u.x*u.x + u.y*u.y + u.z*u.z + u.w*u.w
             + p.x*p.x + p.y*p.y + p.z*p.z + p.w*p.w
             + n.x*n.x + n.y*n.y + n.z*n.z + n.w*n.w;
    unsafeAtomicAdd(sq_acc, sq);
}

// ------- SpMM scatter: dst[didx[e]] += vals[e] * src[sidx[e]]  -------
// 16 lanes per edge, float4 per lane -> coalesced along D; atomics land in
// the 192MB L2 (ue+ie = 77MB resident).
__global__ void spmm_scatter(const float* __restrict__ src,
                             float* __restrict__ dst,
                             const int* __restrict__ sidx,
                             const int* __restrict__ didx,
                             const float* __restrict__ vals, int E) {
    int t = blockIdx.x * blockDim.x + threadIdx.x;
    int e = t >> 4;
    if (e >= E) return;
    int q = (t & 15) * 4;
    float v = vals[e];
    const float* s = src + (size_t)sidx[e] * D + q;
    float*       d = dst + (size_t)didx[e] * D + q;
    float4 x = *(const float4*)s;
    unsafeAtomicAdd(d + 0, v * x.x);
    unsafeAtomicAdd(d + 1, v * x.y);
    unsafeAtomicAdd(d + 2, v * x.z);
    unsafeAtomicAdd(d + 3, v * x.w);
}

// ------- acc[b] += table[idx[b]]  -------
__global__ void acc_gather_add(float* __restrict__ acc,
                               const float* __restrict__ table,
                               const int* __restrict__ idx, int B) {
    int t = blockIdx.x * blockDim.x + threadIdx.x;
    if (t >= B * QUADS) return;
    int b = t >> 4, q = (t & 15) * 4;
    float4 a = *(float4*)(acc + (size_t)b * D + q);
    float4 s = *(const float4*)(table + (size_t)idx[b] * D + q);
    a.x += s.x; a.y += s.y; a.z += s.z; a.w += s.w;
    *(float4*)(acc + (size_t)b * D + q) = a;
}

// ------- BPR dot via V_WMMA_F32_16X16X4_F32 + log-sigmoid reduce -------
// One wave (32 lanes) per 16-row tile. A = u[16xD], B = (p-n)^T[Dx16],
// accumulate 16 chunks of K=4; diagonal of the 16x16 result is the dot.
// A 16x4 layout: lanes 0-15 -> M=lane, VGPR{0,1}=K{0,1}; lanes 16-31 -> K{2,3}.
// B 4x16 layout mirrors C rows: lanes 0-15 -> K{0,1}, lanes 16-31 -> K{2,3},
// N = lane&15.  Both operands therefore index (row = lane&15, kpair = lane>>4).
__global__ __launch_bounds__(32)
void bpr_loss_wmma(const float* __restrict__ u,
                   const float* __restrict__ p,
                   const float* __restrict__ n,
                   float* __restrict__ loss_acc) {
    const int lane = threadIdx.x;
    const int r    = lane & 15;
    const int half = lane >> 4;
    const int row  = blockIdx.x * 16 + r;
    const float* ur = u + (size_t)row * D;
    const float* pr = p + (size_t)row * D;
    const float* nr = n + (size_t)row * D;

    v8f c = {0.f, 0.f, 0.f, 0.f, 0.f, 0.f, 0.f, 0.f};
#pragma unroll
    for (int k0 = 0; k0 < D; k0 += 4) {
        int kk = k0 + 2 * half;
        v2f a, b;
        a.x = ur[kk];
        a.y = ur[kk + 1];
        b.x = pr[kk]     - nr[kk];
        b.y = pr[kk + 1] - nr[kk + 1];
        // 8 args: (neg_a, A, neg_b, B, c_mod, C, reuse_a, reuse_b)
        c = __builtin_amdgcn_wmma_f32_16x16x4_f32(
                false, a, false, b, (short)0, c, false, false);
    }

    // D layout: VGPR i, lanes 0-15 -> M=i, N=lane; lanes 16-31 -> M=i+8, N=lane-16
    __shared__ float tile[16][17];
#pragma unroll
    for (int i = 0; i < 8; ++i)
        tile[i + 8 * half][r] = c[i];
    __syncthreads();

    if (lane < 16) {
        float x = tile[lane][lane] * (1.0f / 16.0f);  // (acc/4)·(acc/4)
        float ls = fminf(x, 0.0f) - log1pf(expf(-fabsf(x)));  // log_sigmoid
        unsafeAtomicAdd(loss_acc, ls);
    }
}

__global__ void finalize(float* __restrict__ out, const float* __restrict__ s,
                         float invU) {
    // s[0]=sum log_sigmoid, s[1]=sum squares
    out[0] = -s[0] + 1e-4f * 0.5f * s[1] * invU;
}

extern "C" void kernel_launch(void* const* d_in, const int* in_sizes, int n_in,
                              void* d_out, int out_size, void* d_ws, size_t ws_size,
                              hipStream_t stream) {
    const int*   user  = (const int*)  d_in[0];
    const int*   pos   = (const int*)  d_in[1];
    const int*   neg   = (const int*)  d_in[2];
    const float* utab  = (const float*)d_in[3];
    const float* itab  = (const float*)d_in[4];
    const int*   eu    = (const int*)  d_in[5];
    const int*   ei    = (const int*)  d_in[6];
    const float* evals = (const float*)d_in[7];

    const int B = in_sizes[0];
    const int U = in_sizes[3] / D;
    const int I = in_sizes[4] / D;
    const int E = in_sizes[5];

    float* ws    = (float*)d_ws;
    float* ue    = ws;                       // U*D
    float* ie    = ue + (size_t)U * D;       // I*D
    float* u_acc = ie + (size_t)I * D;       // B*D
    float* p_acc = u_acc + (size_t)B * D;
    float* n_acc = p_acc + (size_t)B * D;
    float* scal  = n_acc + (size_t)B * D;    // [loss, sq]

    const int TB = 256;
    const int bq_blocks   = (B * QUADS + TB - 1) / TB;
    const int edge_blocks = (E * QUADS + TB - 1) / TB;
    const int zu_blocks   = ((U * D / 4) + TB - 1) / TB;
    const int zi_blocks   = ((I * D / 4) + TB - 1) / TB;

    zero_scalars<<<1, 1, 0, stream>>>(scal);
    init_acc<<<bq_blocks, TB, 0, stream>>>(utab, itab, user, pos, neg,
                                           u_acc, p_acc, n_acc, scal + 1, B);

    const float* ie_src = itab;
    for (int layer = 0; layer < 3; ++layer) {
        zero_f32<<<zu_blocks, TB, 0, stream>>>(ue, U * D / 4);
        spmm_scatter<<<edge_blocks, TB, 0, stream>>>(ie_src, ue, ei, eu, evals, E);
        acc_gather_add<<<bq_blocks, TB, 0, stream>>>(u_acc, ue, user, B);

        zero_f32<<<zi_blocks, TB, 0, stream>>>(ie, I * D / 4);
        spmm_scatter<<<edge_blocks, TB, 0, stream>>>(ue, ie, eu, ei, evals, E);
        acc_gather_add<<<bq_blocks, TB, 0, stream>>>(p_acc, ie, pos, B);
        acc_gather_add<<<bq_blocks, TB, 0, stream>>>(n_acc, ie, neg, B);
        ie_src = ie;
    }

    bpr_loss_wmma<<<B / 16, 32, 0, stream>>>(u_acc, p_acc, n_acc, scal);
    finalize<<<1, 1, 0, stream>>>((float*)d_out, scal, 1.0f / (float)U);
}